// FPNRoIPool_841813590619
// MI455X (gfx1250) — compile-verified
//
#include <hip/hip_runtime.h>
#include <math.h>
#include <stdint.h>

#define POOL   7
#define MAXBIN 12
#define NCH    256

// One block per ROI, 256 threads (8 wave32s), lane -> channel.
// Results staged in LDS, drained with CDNA5 GLOBAL_STORE_ASYNC_FROM_LDS_B128.
__global__ __launch_bounds__(256) void fpn_roipool_kernel(
    const float* __restrict__ f0, const float* __restrict__ f1,
    const float* __restrict__ f2, const float* __restrict__ f3,
    const float* __restrict__ rois, float* __restrict__ out)
{
    __shared__ __align__(16) float slab[NCH * POOL * POOL]; // 50176 B

    const int r = blockIdx.x;
    const int c = threadIdx.x; // channel

    // ROI descriptor (uniform across block -> scalar loads)
    const float rb  = rois[r * 5 + 0];
    const float rx1 = rois[r * 5 + 1];
    const float ry1 = rois[r * 5 + 2];
    const float rx2 = rois[r * 5 + 3];
    const float ry2 = rois[r * 5 + 4];

    // FPN level selection: clip(floor(2 + log2(sqrt(w*h)/224)), 0, 3)
    const float wf = rx2 - rx1 + 1.0f;
    const float hf = ry2 - ry1 + 1.0f;
    int lvl = (int)floorf(2.0f + log2f(sqrtf(wf * hf) / 224.0f));
    lvl = lvl < 0 ? 0 : (lvl > 3 ? 3 : lvl);

    const float* fmap;
    int H, W;
    float scale;
    switch (lvl) {
        case 0:  fmap = f0; H = 200; W = 200; scale = 0.25f;    break;
        case 1:  fmap = f1; H = 100; W = 100; scale = 0.125f;   break;
        case 2:  fmap = f2; H = 50;  W = 50;  scale = 0.0625f;  break;
        default: fmap = f3; H = 25;  W = 25;  scale = 0.03125f; break;
    }

    const int b  = (int)rb;
    const int x1 = (int)floorf(rx1 * scale + 0.5f);
    const int y1 = (int)floorf(ry1 * scale + 0.5f);
    const int x2 = (int)floorf(rx2 * scale + 0.5f);
    const int y2 = (int)floorf(ry2 * scale + 0.5f);
    const int rw = max(x2 - x1 + 1, 1);
    const int rh = max(y2 - y1 + 1, 1);
    const float bh = (float)rh / 7.0f;
    const float bw = (float)rw / 7.0f;

    int hs[POOL], he[POOL], ws[POOL], we[POOL];
#pragma unroll
    for (int p = 0; p < POOL; ++p) {
        int s = (int)floorf((float)p * bh) + y1;
        int e = (int)ceilf((float)(p + 1) * bh) + y1;
        hs[p] = min(max(s, 0), H);
        he[p] = min(max(e, 0), H);
        s = (int)floorf((float)p * bw) + x1;
        e = (int)ceilf((float)(p + 1) * bw) + x1;
        ws[p] = min(max(s, 0), W);
        we[p] = min(max(e, 0), W);
    }

    // Per-lane channel plane; lane streams contiguous w within each row,
    // so every fetched cacheline is fully consumed by its lane.
    const float* base = fmap + (size_t)(b * NCH + c) * (size_t)H * (size_t)W;

    for (int p = 0; p < POOL; ++p) {
        const int h0 = hs[p];
        const int h1 = min(he[p], h0 + MAXBIN); // reference's MB=12 clamp
        const bool vh = he[p] > h0;
        for (int q = 0; q < POOL; ++q) {
            const int w0 = ws[q];
            const int w1 = min(we[q], w0 + MAXBIN);
            float m;
            if (vh && (we[q] > w0)) {
                m = -INFINITY;
                for (int h = h0; h < h1; ++h) {
                    const float* row = base + (size_t)h * W;
                    if (h + 1 < h1)
                        __builtin_prefetch(row + W + w0, 0, 3); // global_prefetch next row
                    for (int wx = w0; wx < w1; ++wx)
                        m = fmaxf(m, row[wx]);
                }
            } else {
                m = 0.0f; // empty bin
            }
            slab[c * (POOL * POOL) + p * POOL + q] = m;
        }
    }

    __syncthreads();

    // Coalesced drain: LDS -> global via async B128 stores (ASYNCcnt).
    float* oslab = out + (size_t)r * (NCH * POOL * POOL);
    const int n128 = (NCH * POOL * POOL) / 4; // 3136 x 16B
    for (int i = c; i < n128; i += 256) {
        unsigned long long gaddr = (unsigned long long)(uintptr_t)(oslab + (size_t)i * 4);
        unsigned int       loff  = (unsigned int)(uintptr_t)&slab[i * 4]; // low 32b = LDS offset
        asm volatile("global_store_async_from_lds_b128 %0, %1, off"
                     :: "v"(gaddr), "v"(loff)
                     : "memory");
    }
    asm volatile("s_wait_asynccnt 0" ::: "memory");
}

extern "C" void kernel_launch(void* const* d_in, const int* in_sizes, int n_in,
                              void* d_out, int out_size, void* d_ws, size_t ws_size,
                              hipStream_t stream) {
    (void)n_in; (void)out_size; (void)d_ws; (void)ws_size;
    const float* f0   = (const float*)d_in[0];
    const float* f1   = (const float*)d_in[1];
    const float* f2   = (const float*)d_in[2];
    const float* f3   = (const float*)d_in[3];
    const float* rois = (const float*)d_in[4];
    const int n_rois = in_sizes[4] / 5;

    fpn_roipool_kernel<<<dim3(n_rois), dim3(256), 0, stream>>>(
        f0, f1, f2, f3, rois, (float*)d_out);
}